// DynamicSequenceChunker_48137993453699
// MI455X (gfx1250) — compile-verified
//
#include <hip/hip_runtime.h>
#include <math.h>

typedef __attribute__((ext_vector_type(2))) float v2f;
typedef __attribute__((ext_vector_type(8))) float v8f;

#define L_SEQ 4096
#define D_DIM 1024
#define B_BATCH 8
#define M_TOT (B_BATCH * L_SEQ)          // 32768
#define OUT_HALF ((size_t)M_TOT * D_DIM) // 33554432
#define SEG_LEN 512
#define N_SEG (L_SEQ / SEG_LEN)          // 8

// ---------------------------------------------------------------------------
// CDNA5 async global->LDS copy (bypasses VGPRs; tracked by ASYNCcnt).
// LDS destination is the offset in lds_off; flat-cast low 32 bits of a __shared__
// pointer give exactly that offset (LDS aperture: addr[31:0] = LDS offset).
// ---------------------------------------------------------------------------
__device__ __forceinline__ void async_ld_b128(uint32_t lds_off, uint64_t gaddr)
{
    asm volatile("global_load_async_to_lds_b128 %0, %1, off"
                 :: "v"(lds_off), "v"(gaddr) : "memory");
}
#define WAIT_ASYNC_LE6() asm volatile("s_wait_asynccnt 0x6" ::: "memory")
#define WAIT_ASYNC_0()   asm volatile("s_wait_asynccnt 0x0" ::: "memory")

// ---------------------------------------------------------------------------
// f32 GEMM via V_WMMA_F32_16X16X4_F32, double-buffered LDS with async copies.
// Block tile: 64 (M) x 128 (N), 256 threads = 8 waves (2x4), each wave a
// 32x32 sub-tile (2x2 WMMA accumulators). Each wave issues 6 async b128 ops
// per k-tile, so ASYNCcnt<=6 == "previous tile landed, next still in flight".
// ---------------------------------------------------------------------------
__device__ __forceinline__ void stage_tile(const float* __restrict__ A,
                                           const float* __restrict__ B,
                                           int K, int ldb, int m0, int n0, int kb,
                                           float (*As)[36], float (*Bs)[128],
                                           int ar, int ac, int br, int bc)
{
    async_ld_b128((uint32_t)(uintptr_t)&As[ar][ac],
                  (uint64_t)(uintptr_t)&A[(size_t)(m0 + ar) * K + kb + ac]);
    async_ld_b128((uint32_t)(uintptr_t)&As[ar + 32][ac],
                  (uint64_t)(uintptr_t)&A[(size_t)(m0 + ar + 32) * K + kb + ac]);
    #pragma unroll
    for (int rr = 0; rr < 4; ++rr)
        async_ld_b128((uint32_t)(uintptr_t)&Bs[br + 8*rr][bc],
                      (uint64_t)(uintptr_t)&B[(size_t)(kb + br + 8*rr) * ldb + n0 + bc]);
}

__global__ __launch_bounds__(256)
void gemm_wmma_f32(const float* __restrict__ A, const float* __restrict__ B,
                   float* __restrict__ C, const float* __restrict__ bias,
                   int K, int ldb, int ldc)
{
    __shared__ float As[2][64][36];    // +4 pad: conflict-free b64 frag reads
    __shared__ float Bs[2][32][128];

    const int t    = threadIdx.x;
    const int lane = t & 31;
    const int wave = t >> 5;
    const int wm   = wave >> 2;        // 0..1
    const int wn   = wave & 3;         // 0..3
    const int lr   = lane & 15;
    const int kh   = lane >> 4;        // 0..1
    const int m0   = blockIdx.y * 64;
    const int n0   = blockIdx.x * 128;

    const v8f vzero = {0.f,0.f,0.f,0.f,0.f,0.f,0.f,0.f};
    v8f acc[2][2];
    acc[0][0] = vzero; acc[0][1] = vzero; acc[1][0] = vzero; acc[1][1] = vzero;

    const int ar = t >> 3;            // A fill rows ar, ar+32 (float4 cols)
    const int ac = (t & 7) * 4;
    const int br = t >> 5;            // B fill rows br+{0,8,16,24}
    const int bc = (t & 31) * 4;

    stage_tile(A, B, K, ldb, m0, n0, 0, As[0], Bs[0], ar, ac, br, bc);

    const int numK = K / 32;
    for (int kt = 0; kt < numK; ++kt) {
        const int pb = kt & 1;
        if (kt + 1 < numK) {
            stage_tile(A, B, K, ldb, m0, n0, (kt + 1) * 32,
                       As[pb ^ 1], Bs[pb ^ 1], ar, ac, br, bc);
            WAIT_ASYNC_LE6();          // tile kt landed; kt+1 in flight
        } else {
            WAIT_ASYNC_0();
        }
        __syncthreads();

        #pragma unroll
        for (int kk = 0; kk < 32; kk += 4) {
            v2f a0 = *(const v2f*)&As[pb][wm*32 +  0 + lr][kk + 2*kh];
            v2f a1 = *(const v2f*)&As[pb][wm*32 + 16 + lr][kk + 2*kh];
            v2f b0, b1;
            b0.x = Bs[pb][kk + 2*kh + 0][wn*32 +  0 + lr];
            b0.y = Bs[pb][kk + 2*kh + 1][wn*32 +  0 + lr];
            b1.x = Bs[pb][kk + 2*kh + 0][wn*32 + 16 + lr];
            b1.y = Bs[pb][kk + 2*kh + 1][wn*32 + 16 + lr];
            acc[0][0] = __builtin_amdgcn_wmma_f32_16x16x4_f32(false, a0, false, b0, (short)0, acc[0][0], false, false);
            acc[0][1] = __builtin_amdgcn_wmma_f32_16x16x4_f32(false, a0, false, b1, (short)0, acc[0][1], false, false);
            acc[1][0] = __builtin_amdgcn_wmma_f32_16x16x4_f32(false, a1, false, b0, (short)0, acc[1][0], false, false);
            acc[1][1] = __builtin_amdgcn_wmma_f32_16x16x4_f32(false, a1, false, b1, (short)0, acc[1][1], false, false);
        }
        __syncthreads();               // all waves done reading before overwrite
    }

    // store: lane holds (m = r + 8*kh, n = lr) of each 16x16 tile
    #pragma unroll
    for (int mi = 0; mi < 2; ++mi) {
        #pragma unroll
        for (int ni = 0; ni < 2; ++ni) {
            const int col = n0 + wn*32 + ni*16 + lr;
            const float bv = bias ? bias[col] : 0.0f;
            #pragma unroll
            for (int r = 0; r < 8; ++r) {
                const int row = m0 + wm*32 + mi*16 + r + 8*kh;
                C[(size_t)row * ldc + col] = acc[mi][ni][r] + bv;
            }
        }
    }
}

// ---------------------------------------------------------------------------
// probs[m] = (1 - cos(q_m, ks_m)) / 2 ; one wave32 per token.
// ---------------------------------------------------------------------------
__global__ __launch_bounds__(256)
void probs_kernel(const float* __restrict__ qk, const float* __restrict__ start_key,
                  float* __restrict__ probs)
{
    const int m    = blockIdx.x * 8 + (threadIdx.x >> 5);
    const int lane = threadIdx.x & 31;
    const int l    = m & (L_SEQ - 1);
    const float* q  = qk + (size_t)m * 2048;
    const float* ks = (l == 0) ? start_key : qk + (size_t)(m - 1) * 2048 + 1024;

    float dq = 0.f, dk = 0.f, dd = 0.f;
    for (int i = lane; i < D_DIM; i += 32) {
        float qa = q[i], ka = ks[i];
        dq += qa * qa;
        dk += ka * ka;
        dd += qa * ka;
    }
    #pragma unroll
    for (int off = 16; off > 0; off >>= 1) {
        dq += __shfl_xor(dq, off, 32);
        dk += __shfl_xor(dk, off, 32);
        dd += __shfl_xor(dd, off, 32);
    }
    if (lane == 0) {
        float denom = fmaxf(sqrtf(dq) * sqrtf(dk), 1e-8f);
        probs[m] = (1.0f - dd / denom) * 0.5f;
    }
}

// ---------------------------------------------------------------------------
// Per-batch: boundaries, cumsum -> chunk_id, slot scatter, segmented gate
// prefix products, aux partials. One 1024-thread block per batch.
// ---------------------------------------------------------------------------
__global__ __launch_bounds__(1024)
void chunk_kernel(const float* __restrict__ probs, int* __restrict__ chunk_id,
                  int* __restrict__ srcpos, float* __restrict__ bprob_slot,
                  float* __restrict__ gprod, int* __restrict__ num_chunks,
                  float* __restrict__ auxpart)
{
    __shared__ int   si[1024];
    __shared__ float sf[1024];
    const int b = blockIdx.x, t = threadIdx.x;

    float p[4]; int bnd[4], cum[4];
    int run = 0; float psum = 0.f;
    #pragma unroll
    for (int j = 0; j < 4; ++j) {
        const int l = t * 4 + j;
        p[j]   = probs[b * L_SEQ + l];
        bnd[j] = (l == 0) || (p[j] > 0.5f);
        run   += bnd[j];
        cum[j] = run;
        psum  += p[j];
    }
    si[t] = run; sf[t] = psum;
    __syncthreads();
    for (int off = 1; off < 1024; off <<= 1) {         // inclusive scan
        int add = (t >= off) ? si[t - off] : 0;
        __syncthreads();
        si[t] += add;
        __syncthreads();
    }
    const int excl = si[t] - run;
    const int nc   = si[1023];
    #pragma unroll
    for (int j = 0; j < 4; ++j) {
        const int l   = t * 4 + j;
        const int cid = excl + cum[j] - 1;
        chunk_id[b * L_SEQ + l] = cid;
        if (bnd[j]) {
            srcpos[b * L_SEQ + cid]     = l;
            bprob_slot[b * L_SEQ + cid] = p[j];
        }
    }
    __syncthreads();                                   // slots fully written
    // inclusive gate products within each 512-slot segment (gate=1 past nc)
    if (t < N_SEG) {
        float r = 1.0f;
        for (int j = 0; j < SEG_LEN; ++j) {
            const int s = t * SEG_LEN + j;
            const float g = (s < nc) ? (1.0f - bprob_slot[b * L_SEQ + s]) : 1.0f;
            r *= g;
            gprod[b * L_SEQ + s] = r;
        }
    }
    for (int off = 512; off > 0; off >>= 1) {          // reduce sum(probs)
        if (t < off) sf[t] += sf[t + off];
        __syncthreads();
    }
    if (t == 0) {
        num_chunks[b] = nc;
        const float F = (float)nc / (float)L_SEQ;      // straight-through fwd
        const float G = sf[0] / (float)L_SEQ;
        auxpart[b] = 1.2f * (5.0f * F * G + (1.0f - F) * (1.0f - G));
    }
}

// ---------------------------------------------------------------------------
// downsampled[b,s,:] = tokens[b, srcpos, :] * bprob  (zeros for padded slots)
// ---------------------------------------------------------------------------
__global__ __launch_bounds__(256)
void down_kernel(const float* __restrict__ tokens, const int* __restrict__ srcpos,
                 const float* __restrict__ bprob_slot, const int* __restrict__ num_chunks,
                 float* __restrict__ down)
{
    const int bs = blockIdx.x;              // b*L + s
    const int b  = bs >> 12;
    const int s  = bs & (L_SEQ - 1);
    float4 v = {0.f, 0.f, 0.f, 0.f};
    if (s < num_chunks[b]) {
        const int   l = srcpos[bs];
        const float w = bprob_slot[bs];
        float4 tk = ((const float4*)tokens)[((size_t)(b * L_SEQ + l)) * 256 + threadIdx.x];
        v.x = tk.x * w; v.y = tk.y * w; v.z = tk.z * w; v.w = tk.w * w;
    }
    ((float4*)down)[(size_t)bs * 256 + threadIdx.x] = v;
}

// ---------------------------------------------------------------------------
// Segment-local gated scans (h_in = 0): h_s = g_s h_{s-1} + x_s.
// True value = local + gprod * H_in(segment); fixup applied in ups_kernel.
// One block per (batch, 256-ch group, segment): 256 blocks.
// ---------------------------------------------------------------------------
__global__ __launch_bounds__(256)
void seg_scan_kernel(const float* __restrict__ down, const float* __restrict__ bprob_slot,
                     const int* __restrict__ num_chunks, float* __restrict__ scanned,
                     float* __restrict__ h_end)
{
    const int x   = blockIdx.x;
    const int b   = x >> 5;
    const int g   = (x >> 3) & 3;
    const int seg = x & 7;
    const int d   = g * 256 + threadIdx.x;
    const int nc  = num_chunks[b];
    const int s0  = seg * SEG_LEN;
    const int s1  = (s0 + SEG_LEN < nc) ? (s0 + SEG_LEN) : nc;
    float h = 0.f;
    for (int s = s0; s < s1; ++s) {
        const float gate = 1.0f - bprob_slot[b * L_SEQ + s];
        const size_t idx = ((size_t)(b * L_SEQ + s)) * D_DIM + d;
        h = gate * h + down[idx];
        scanned[idx] = h;
    }
    // past nc: gate=1, x=0 -> h unchanged (consistent with padded gprod)
    h_end[((size_t)(b * N_SEG + seg)) * D_DIM + d] = h;
}

// ---------------------------------------------------------------------------
// Compose segment carries: H_in(0)=0; H_in(s+1) = h_end(s) + A(s)*H_in(s),
// A(s) = gprod at segment end. 32 blocks x 256 threads.
// ---------------------------------------------------------------------------
__global__ __launch_bounds__(256)
void seg_compose_kernel(const float* __restrict__ h_end, const float* __restrict__ gprod,
                        float* __restrict__ H_in)
{
    const int b = blockIdx.x >> 2;
    const int d = (blockIdx.x & 3) * 256 + threadIdx.x;
    float h = 0.f;
    #pragma unroll
    for (int seg = 0; seg < N_SEG; ++seg) {
        H_in[((size_t)(b * N_SEG + seg)) * D_DIM + d] = h;
        const float A = gprod[b * L_SEQ + seg * SEG_LEN + SEG_LEN - 1];
        h = h_end[((size_t)(b * N_SEG + seg)) * D_DIM + d] + A * h;
    }
}

// ---------------------------------------------------------------------------
// ups[b,l,:] += scanned[b,cid,:] + gprod[b,cid] * H_in[b,seg(cid),:]
// (residual projection already stored in ups by GEMM2)
// ---------------------------------------------------------------------------
__global__ __launch_bounds__(256)
void ups_kernel(const float* __restrict__ scanned, const int* __restrict__ chunk_id,
                const float* __restrict__ gprod, const float* __restrict__ H_in,
                float* __restrict__ ups)
{
    const int bl  = blockIdx.x;
    const int b   = bl >> 12;
    const int cid = chunk_id[bl];
    const int seg = cid >> 9;
    const float c = gprod[b * L_SEQ + cid];
    const size_t src = ((size_t)(b * L_SEQ + cid)) * 256 + threadIdx.x;
    const size_t hsr = ((size_t)(b * N_SEG + seg)) * 256 + threadIdx.x;
    const size_t dst = (size_t)bl * 256 + threadIdx.x;
    float4 s4 = ((const float4*)scanned)[src];
    float4 h4 = ((const float4*)H_in)[hsr];
    float4 v  = ((float4*)ups)[dst];
    v.x += s4.x + c * h4.x;
    v.y += s4.y + c * h4.y;
    v.z += s4.z + c * h4.z;
    v.w += s4.w + c * h4.w;
    ((float4*)ups)[dst] = v;
}

__global__ void finalize_aux(const float* __restrict__ auxpart, float* __restrict__ out)
{
    float s = 0.f;
    for (int i = 0; i < B_BATCH; ++i) s += auxpart[i];
    *out = (s / (float)B_BATCH) * 0.03f;
}

// ---------------------------------------------------------------------------
extern "C" void kernel_launch(void* const* d_in, const int* in_sizes, int n_in,
                              void* d_out, int out_size, void* d_ws, size_t ws_size,
                              hipStream_t stream)
{
    (void)in_sizes; (void)n_in; (void)out_size; (void)ws_size;
    const float* tokens    = (const float*)d_in[0];  // [8,4096,1024]
    const float* W_qk      = (const float*)d_in[1];  // [1024,2048]
    const float* start_key = (const float*)d_in[2];  // [1024]
    const float* W_res     = (const float*)d_in[3];  // [1024,1024]
    const float* b_res     = (const float*)d_in[4];  // [1024]

    float* out      = (float*)d_out;
    float* down_out = out;                 // output 0: [8,4096,1024]
    float* ups_out  = out + OUT_HALF;      // output 1: [8,4096,1024]
    float* aux_out  = out + 2 * OUT_HALF;  // output 2: scalar
    float* qk       = out;                 // scratch alias (overwritten later)

    char* w = (char*)d_ws;
    float* scanned    = (float*)(w);                 // 134217728 B
    float* probs      = (float*)(w + 134217728);     // 131072 B
    int*   chunk_id   = (int*)  (w + 134348800);     // 131072 B
    int*   srcpos     = (int*)  (w + 134479872);     // 131072 B
    float* bprob_slot = (float*)(w + 134610944);     // 131072 B
    float* gprod      = (float*)(w + 134742016);     // 131072 B
    float* h_end      = (float*)(w + 134873088);     // 262144 B
    float* H_in       = (float*)(w + 135135232);     // 262144 B
    int*   num_chunks = (int*)  (w + 135397376);     // 32 B
    float* auxpart    = (float*)(w + 135397408);     // 32 B

    // 1) qk = tokens @ W_qk  -> d_out scratch ([32768,2048])
    gemm_wmma_f32<<<dim3(2048 / 128, M_TOT / 64), 256, 0, stream>>>(
        tokens, W_qk, qk, nullptr, D_DIM, 2048, 2048);
    // 2) per-token cosine -> probs
    probs_kernel<<<M_TOT / 8, 256, 0, stream>>>(qk, start_key, probs);
    // 3) boundaries / chunk ids / gate products / aux
    chunk_kernel<<<B_BATCH, 1024, 0, stream>>>(probs, chunk_id, srcpos, bprob_slot,
                                               gprod, num_chunks, auxpart);
    // 4) residual projection (+bias) into ups output region
    gemm_wmma_f32<<<dim3(1024 / 128, M_TOT / 64), 256, 0, stream>>>(
        tokens, W_res, ups_out, b_res, D_DIM, 1024, 1024);
    // 5) compact boundary tokens (zero-pads invalid slots)
    down_kernel<<<M_TOT, 256, 0, stream>>>(tokens, srcpos, bprob_slot, num_chunks, down_out);
    // 6) segmented gated scans + carry composition
    seg_scan_kernel<<<B_BATCH * 4 * N_SEG, 256, 0, stream>>>(down_out, bprob_slot,
                                                             num_chunks, scanned, h_end);
    seg_compose_kernel<<<B_BATCH * 4, 256, 0, stream>>>(h_end, gprod, H_in);
    // 7) gather + carry fixup, add to residual
    ups_kernel<<<M_TOT, 256, 0, stream>>>(scanned, chunk_id, gprod, H_in, ups_out);
    // 8) aux scalar
    finalize_aux<<<1, 1, 0, stream>>>(auxpart, aux_out);
}